// SpatialGraphConvLayer_22548578304757
// MI455X (gfx1250) — compile-verified
//
#include <hip/hip_runtime.h>

typedef __attribute__((ext_vector_type(2))) float v2f;
typedef __attribute__((ext_vector_type(8))) float v8f;

#define N_NODES    8192
#define F_OUT      64
#define MASK_WORDS (N_NODES / 32)   // 256 dwords per row
#define CHUNK_J    128              // j rows staged in LDS per buffer (32 KB)
#define CHUNK_BYTES (CHUNK_J * F_OUT * 4)
#define TILE_I     32               // i rows per workgroup (8 waves x 4)

// ---------------------------------------------------------------------------
// Kernel 1: compact adj (256 MB int32) -> 8 MB bitmask + inverse degree.
// One wave32 per row; ballot turns 32 lane-predicates into one mask dword.
// Streams adj exactly once at HBM rate (~11 us floor at 23.3 TB/s).
// ---------------------------------------------------------------------------
__global__ void __launch_bounds__(256)
pack_mask_kernel(const int* __restrict__ adj,
                 unsigned* __restrict__ maskw,
                 float* __restrict__ inv_cnt) {
    const int wave = (int)((blockIdx.x * blockDim.x + threadIdx.x) >> 5);
    const int lane = (int)(threadIdx.x & 31);
    if (wave >= N_NODES) return;
    const int* row = adj + (size_t)wave * N_NODES;
    unsigned cnt = 0;
#pragma unroll 4
    for (int w = 0; w < MASK_WORDS; ++w) {
        int a = row[w * 32 + lane];
        unsigned m = __builtin_amdgcn_ballot_w32(a != 0);
        if (lane == 0) maskw[(size_t)wave * MASK_WORDS + w] = m;
        cnt += (unsigned)__popc(m);
    }
    if (lane == 0) inv_cnt[wave] = cnt ? 1.0f / (float)cnt : 0.0f;
}

// ---------------------------------------------------------------------------
// Kernel 2: projb = coords @ U + b via V_WMMA_F32_16X16X4_F32 (exact f32).
// One wave per 16x16 output tile; layouts per CDNA5 ISA 7.12.2.
// ---------------------------------------------------------------------------
__global__ void __launch_bounds__(128)
proj_wmma_kernel(const float* __restrict__ coords,
                 const float* __restrict__ U,
                 const float* __restrict__ bias,
                 float* __restrict__ projb) {
    const int tile  = (int)(blockIdx.x * 4 + (threadIdx.x >> 5)); // 2048 tiles
    const int mtile = tile >> 2;
    const int ntile = tile & 3;
    const int lane  = (int)(threadIdx.x & 31);
    const int half  = lane >> 4;
    const int l16   = lane & 15;

    v2f a, bm;
    if (half == 0) {
        const float* cp = coords + (size_t)(mtile * 16 + l16) * 2;
        a.x  = cp[0];                             // A: K=0
        a.y  = cp[1];                             // A: K=1
        bm.x = U[0 * F_OUT + ntile * 16 + l16];   // B row K=0
        bm.y = U[1 * F_OUT + ntile * 16 + l16];   // B row K=1
    } else {
        a.x = 0.f; a.y = 0.f;                     // K=2,3 unused
        bm.x = 0.f; bm.y = 0.f;
    }

    v8f c = {};
    c = __builtin_amdgcn_wmma_f32_16x16x4_f32(
            false, a, false, bm, (short)0, c, false, false);

    const int   col  = ntile * 16 + l16;
    const float bcol = bias[col];
#pragma unroll
    for (int r = 0; r < 8; ++r) {
        const int row = mtile * 16 + r + half * 8;
        projb[(size_t)row * F_OUT + col] = c[r] + bcol;
    }
}

// ---------------------------------------------------------------------------
// Kernel 3: masked relu mean. Block = 8 waves, TILE_I=32 (4 i's per wave),
// lanes <-> f-pairs. projb chunks are double-buffered in LDS via
// GLOBAL_LOAD_ASYNC_TO_LDS_B128 (ASYNCcnt-tracked DMA, overlapped with
// compute). Protocol per chunk c:
//   1) s_wait_asynccnt 0      -> this wave's chunk-c slice landed in LDS
//   2) __syncthreads()        -> everyone's slice landed; everyone finished
//                                reading buffer[(c+1)&1] during chunk c-1
//   3) issue async copy of chunk c+1 into buffer[(c+1)&1]   (overlaps 4)
//   4) compute chunk c from buffer[c&1]
// Taken-path math per (i,j): pk add + 2x max + pk add (base_i = b - projb_i
// pre-folded); zero mask bits skipped by wave-uniform scalar branches.
// ---------------------------------------------------------------------------
__global__ void __launch_bounds__(256, 2)
spatial_main_kernel(const float* __restrict__ projb,
                    const unsigned* __restrict__ maskw,
                    const float* __restrict__ inv_cnt,
                    const float* __restrict__ bias,
                    float* __restrict__ out) {
    __shared__ float lds_pb[2 * CHUNK_J * F_OUT];   // 64 KB double buffer @ LDS 0

    const int t    = (int)threadIdx.x;
    const int lane = t & 31;
    const int wv   = __builtin_amdgcn_readfirstlane(t >> 5);  // uniform 0..7
    const int i0   = (int)blockIdx.x * TILE_I + wv * 4;

    float2 bl = *(const float2*)(bias + 2 * lane);

    float2 base[4], acc[4];
#pragma unroll
    for (int k = 0; k < 4; ++k) {
        float2 pbi = *(const float2*)(projb + (size_t)(i0 + k) * F_OUT + 2 * lane);
        base[k].x = bl.x - pbi.x;
        base[k].y = bl.y - pbi.y;
        acc[k].x = 0.f; acc[k].y = 0.f;
    }

    // prologue: async-copy chunk 0 into buffer 0 (16B per lane per issue)
    {
        unsigned vdst  = (unsigned)t * 16u;
        unsigned vaddr = (unsigned)t * 16u;
#pragma unroll
        for (int k = 0; k < 8; ++k)
            asm volatile("global_load_async_to_lds_b128 %0, %1, %2 offset:%3"
                         :: "v"(vdst), "v"(vaddr), "s"(projb), "n"(k * 4096)
                         : "memory");
    }

    const int nchunk = N_NODES / CHUNK_J;   // 64
    for (int c = 0; c < nchunk; ++c) {
        asm volatile("s_wait_asynccnt 0x0" ::: "memory");
        __syncthreads();

        if (c + 1 < nchunk) {
            unsigned vdst  = (unsigned)(((c + 1) & 1) * CHUNK_BYTES) + (unsigned)t * 16u;
            unsigned vaddr = (unsigned)((c + 1) * CHUNK_BYTES) + (unsigned)t * 16u;
#pragma unroll
            for (int k = 0; k < 8; ++k)
                asm volatile("global_load_async_to_lds_b128 %0, %1, %2 offset:%3"
                             :: "v"(vdst), "v"(vaddr), "s"(projb), "n"(k * 4096)
                             : "memory");
        }

        const float* cur = lds_pb + (c & 1) * (CHUNK_J * F_OUT);

        // 4 mask dwords per i for this chunk (uniform -> scalar loads)
        unsigned mw[4][CHUNK_J / 32];
#pragma unroll
        for (int k = 0; k < 4; ++k)
#pragma unroll
            for (int wd = 0; wd < CHUNK_J / 32; ++wd)
                mw[k][wd] = maskw[(size_t)(i0 + k) * MASK_WORDS + c * (CHUNK_J / 32) + wd];

#pragma unroll 1
        for (int wd = 0; wd < CHUNK_J / 32; ++wd) {
            const unsigned m0 = mw[0][wd], m1 = mw[1][wd];
            const unsigned m2 = mw[2][wd], m3 = mw[3][wd];
#pragma unroll
            for (int b = 0; b < 32; ++b) {
                const int jl = wd * 32 + b;
                float2 pbv = *(const float2*)(cur + jl * F_OUT + 2 * lane);
                const unsigned bit = 1u << b;
                if (m0 & bit) { acc[0].x += fmaxf(pbv.x + base[0].x, 0.f);
                                acc[0].y += fmaxf(pbv.y + base[0].y, 0.f); }
                if (m1 & bit) { acc[1].x += fmaxf(pbv.x + base[1].x, 0.f);
                                acc[1].y += fmaxf(pbv.y + base[1].y, 0.f); }
                if (m2 & bit) { acc[2].x += fmaxf(pbv.x + base[2].x, 0.f);
                                acc[2].y += fmaxf(pbv.y + base[2].y, 0.f); }
                if (m3 & bit) { acc[3].x += fmaxf(pbv.x + base[3].x, 0.f);
                                acc[3].y += fmaxf(pbv.y + base[3].y, 0.f); }
            }
        }
    }

#pragma unroll
    for (int k = 0; k < 4; ++k) {
        const float ic = inv_cnt[i0 + k];   // 0 when no neighbors -> output 0
        float2 r; r.x = acc[k].x * ic; r.y = acc[k].y * ic;
        *(float2*)(out + (size_t)(i0 + k) * F_OUT + 2 * lane) = r;
    }
}

// ---------------------------------------------------------------------------
extern "C" void kernel_launch(void* const* d_in, const int* in_sizes, int n_in,
                              void* d_out, int out_size, void* d_ws, size_t ws_size,
                              hipStream_t stream) {
    // inputs: x (unused), adj int32, coords f32, U f32, b f32
    const int*   adj    = (const int*)d_in[1];
    const float* coords = (const float*)d_in[2];
    const float* U      = (const float*)d_in[3];
    const float* bias   = (const float*)d_in[4];
    float*       out    = (float*)d_out;

    char* ws = (char*)d_ws;
    float*    projb  = (float*)ws;     ws += (size_t)N_NODES * F_OUT * sizeof(float);          // 2 MB
    unsigned* maskw  = (unsigned*)ws;  ws += (size_t)N_NODES * MASK_WORDS * sizeof(unsigned);  // 8 MB
    float*    invcnt = (float*)ws;                                                             // 32 KB

    hipLaunchKernelGGL(pack_mask_kernel, dim3(N_NODES / 8), dim3(256), 0, stream,
                       adj, maskw, invcnt);
    hipLaunchKernelGGL(proj_wmma_kernel, dim3(512), dim3(128), 0, stream,
                       coords, U, bias, projb);
    hipLaunchKernelGGL(spatial_main_kernel, dim3(N_NODES / TILE_I), dim3(256), 0, stream,
                       projb, maskw, invcnt, bias, out);
}